// SO3Convolution_58222576665053
// MI455X (gfx1250) — compile-verified
//
#include <hip/hip_runtime.h>
#include <hip/hip_bf16.h>
#include <stdint.h>

typedef __attribute__((ext_vector_type(2))) float v2f;
typedef __attribute__((ext_vector_type(8))) float v8f;

#define N_VOX 1024
#define AB_DIM 4      // A*B = 2*2
#define I_DIM 32
#define O_DIM 32
#define S_TOT 565     // sum of (2l+1)^2 over l in {0,2,4,6,8}

// Contraction index ordering (shared by A and B): c = k*32 + i  (i = c&31, k = c>>5).
// A panel staged into LDS via ASYNC global->LDS copies with per-lane scattered
// dsaddr implementing the skew-2 layout: off(c) = c + 2*(c>>5) = k*34 + i.
// One wave accumulates TWO 16x16 output tiles (two voxel row-tiles, same cols),
// sharing one B fragment per WMMA K-step -> 2 wmma per pair of global loads.
template<int L, int SH_OFF>
__global__ __launch_bounds__(256) void so3conv_wmma(
    const float* __restrict__ data,   // [N, A, B, I, m, m]
    const float* __restrict__ w,      // [A, B, I, O, m, m]
    const float* __restrict__ bias,   // [1, A, B, O, 1, 1]
    float* __restrict__ out)          // [N, A, B, O, S_TOT]
{
    constexpr int m        = 2 * L + 1;
    constexpr int mm       = m * m;
    constexpr int K        = I_DIM * m;        // contraction length
    constexpr int NC       = O_DIM * m;        // output columns (o*m + j)
    constexpr int colTiles = NC / 16;          // = 2m
    constexpr int blocksPerRow = (colTiles + 7) / 8;
    constexpr int ROWS     = 32;               // two 16-row tiles per block
    constexpr int pitch    = 34 * m;           // skewed row pitch (even)

    __shared__ __align__(16) float sA[ROWS * pitch];

    // block id -> (ab, lr, rowBlock, colBlock)
    int bid = blockIdx.x;
    const int colBlock = bid % blocksPerRow;   bid /= blocksPerRow;
    const int rowBlock = bid % (N_VOX / ROWS); bid /= (N_VOX / ROWS);
    const int lr       = bid % m;              bid /= m;
    const int ab       = bid;                  // a*B + b in 0..3

    const int tid  = threadIdx.x;
    const int wave = tid >> 5;
    const int lane = tid & 31;
    const int half = lane >> 4;
    const int r    = lane & 15;

    const int nBase = rowBlock * ROWS;

    // ---- async staging: 32 voxel rows x K -> skewed LDS (no VGPR round-trip) ----
    // data[n, ab, i, lr, k] at n*AB*I*mm + ab*I*mm + i*mm + lr*m + k
    {
        constexpr int dRowStride = AB_DIM * I_DIM * mm;
        const float* dBase = data + (long)nBase * dRowStride + ab * I_DIM * mm + lr * m;
        for (int idx = tid; idx < ROWS * K; idx += 256) {
            const int row = idx / K;
            const int cl  = idx - row * K;     // linear (i,k): i = cl/m, k = cl%m
            const int i   = cl / m;
            const int k   = cl - i * m;
            const float* g = dBase + (long)row * dRowStride + i * mm + k;
            // permuted contraction position c = k*32 + i -> skewed off = k*34 + i
            const unsigned ldsOff =
                (unsigned)(uintptr_t)(&sA[row * pitch + k * 34 + i]);
            asm volatile("global_load_async_to_lds_b32 %0, %1, off"
                         :: "v"(ldsOff),
                            "v"((unsigned long long)(uintptr_t)g)
                         : "memory");
        }
        asm volatile("s_wait_asynccnt 0x0" ::: "memory");
    }
    __syncthreads();

    const int colTile = colBlock * 8 + wave;
    if (colTile < colTiles) {
        const int col = colTile * 16 + r;      // o*m + j (fixed per lane)
        const int o   = col / m;
        const int j   = col - o * m;

        // w[ab, i, o, k, j] at ab*I*O*mm + i*O*mm + o*mm + k*m + j
        constexpr int wIStride = O_DIM * mm;
        const float* wBase = w + ((long)ab * I_DIM * O_DIM + o) * mm + j;

        const float* sRow0 = sA + r * pitch;            // rows 0..15
        const float* sRow1 = sA + (16 + r) * pitch;     // rows 16..31
        v8f acc0 = {}, acc1 = {};

        #pragma unroll 4
        for (int kc = 0; kc < K / 4; ++kc) {
            const int c0 = kc * 4 + 2 * half;  // even -> i0 even, no i-wrap at c0+1
            const int i0 = c0 & 31;
            const int k0 = c0 >> 5;
            const int sOff = c0 + 2 * k0;      // skewed LDS offset (even, 8B aligned)

            const v2f a0 = *reinterpret_cast<const v2f*>(sRow0 + sOff);
            const v2f a1 = *reinterpret_cast<const v2f*>(sRow1 + sOff);

            const float* wp = wBase + i0 * wIStride + k0 * m;
            v2f b;
            b.x = wp[0];           // B[c0,   col]
            b.y = wp[wIStride];    // B[c0+1, col] (i0+1, same k0)

            acc0 = __builtin_amdgcn_wmma_f32_16x16x4_f32(
                       false, a0, false, b, (short)0, acc0, false, false);
            acc1 = __builtin_amdgcn_wmma_f32_16x16x4_f32(
                       false, a1, false, b, (short)0, acc1, false, false);
        }

        float biasv = 0.0f;
        if (L == 0) biasv = bias[ab * O_DIM + o];

        // out[n, ab, o, SH_OFF + lr*m + j]
        const long oColOff = (long)(ab * O_DIM + o) * S_TOT + SH_OFF + lr * m + j;
        constexpr long oRowStride = (long)AB_DIM * O_DIM * S_TOT;
        #pragma unroll
        for (int v = 0; v < 8; ++v) {
            const int row = v + 8 * half;      // C layout: M = v + 8*(lane>=16)
            out[(long)(nBase + row)      * oRowStride + oColOff] = (float)m * acc0[v] + biasv;
            out[(long)(nBase + 16 + row) * oRowStride + oColOff] = (float)m * acc1[v] + biasv;
        }
    }
}

extern "C" void kernel_launch(void* const* d_in, const int* in_sizes, int n_in,
                              void* d_out, int out_size, void* d_ws, size_t ws_size,
                              hipStream_t stream) {
    (void)in_sizes; (void)n_in; (void)out_size; (void)d_ws; (void)ws_size;
    // setup_inputs order: data_l0, w_l0, data_l2, w_l2, ..., data_l8, w_l8, bias
    const float* bias = (const float*)d_in[10];
    float* out = (float*)d_out;
    dim3 blk(256);

    // grid = AB(4) * m * rowBlocks(32) * ceil(2m/8)
    so3conv_wmma<0,   0><<<dim3(4 *  1 * 32 * 1), blk, 0, stream>>>(
        (const float*)d_in[0], (const float*)d_in[1], bias, out);
    so3conv_wmma<2,   1><<<dim3(4 *  5 * 32 * 2), blk, 0, stream>>>(
        (const float*)d_in[2], (const float*)d_in[3], bias, out);
    so3conv_wmma<4,  26><<<dim3(4 *  9 * 32 * 3), blk, 0, stream>>>(
        (const float*)d_in[4], (const float*)d_in[5], bias, out);
    so3conv_wmma<6, 107><<<dim3(4 * 13 * 32 * 4), blk, 0, stream>>>(
        (const float*)d_in[6], (const float*)d_in[7], bias, out);
    so3conv_wmma<8, 276><<<dim3(4 * 17 * 32 * 5), blk, 0, stream>>>(
        (const float*)d_in[8], (const float*)d_in[9], bias, out);
}